// RGCNConv_24790551233453
// MI455X (gfx1250) — compile-verified
//
#include <hip/hip_runtime.h>
#include <hip/hip_bf16.h>

typedef float v2f __attribute__((ext_vector_type(2)));
typedef float v8f __attribute__((ext_vector_type(8)));

#if defined(__gfx1250__)
#if !__has_builtin(__builtin_amdgcn_wmma_f32_16x16x4_f32)
#error "missing __builtin_amdgcn_wmma_f32_16x16x4_f32 on gfx1250"
#endif
#endif

#define DIM 128

// C[M x 128] = A[M x 128] @ W[128 x 128], row-major, f32 WMMA 16x16x4.
// One wave computes two adjacent 16x16 tiles of C (cols [n0,n0+32)), two
// accumulator chains interleaved to hide WMMA latency. 4 waves per block.
__global__ __launch_bounds__(128) void wmma_gemm_f32(
    const float* __restrict__ A, const float* __restrict__ W,
    float* __restrict__ C, int M)
{
    const int wave = threadIdx.x >> 5;
    const int lane = threadIdx.x & 31;
    const int mTiles = M >> 4;
    const int tile = blockIdx.x * 4 + wave;
    if (tile >= mTiles * 4) return;   // wave-uniform exit; EXEC stays all-ones
    const int mt   = tile >> 2;
    const int np   = tile & 3;        // which pair of 16-col tiles
    const int half = lane >> 4;       // 0: K even pair, 1: K odd pair
    const int l15  = lane & 15;
    const int m0   = mt << 4;
    const int n0   = np << 5;

    // A 16x4 layout: lanes 0-15 -> K=0,1 ; lanes 16-31 -> K=2,3 ; M = lane&15
    const float* aptr  = A + (size_t)(m0 + l15) * DIM + (half << 1);
    // B 4x16 layout: VGPR0 = row K (=kk+2*half), VGPR1 = row K+1 ; N = lane&15
    const float* bptr0 = W + (size_t)(half << 1) * DIM + n0 + l15;
    const float* bptr1 = bptr0 + 16;

    v8f acc0 = {};
    v8f acc1 = {};
#pragma unroll
    for (int kk = 0; kk < DIM; kk += 4) {
        v2f a = *(const v2f*)(aptr + kk);
        const float* bk0 = bptr0 + (size_t)kk * DIM;
        const float* bk1 = bptr1 + (size_t)kk * DIM;
        v2f b0, b1;
        b0.x = bk0[0];  b0.y = bk0[DIM];
        b1.x = bk1[0];  b1.y = bk1[DIM];
        acc0 = __builtin_amdgcn_wmma_f32_16x16x4_f32(false, a, false, b0,
                                                     (short)0, acc0, false, false);
        acc1 = __builtin_amdgcn_wmma_f32_16x16x4_f32(false, a, false, b1,
                                                     (short)0, acc1, false, false);
    }
    // C/D layout: acc[j] -> row m0 + j + 8*half, col n0 + (lane&15)
    float* cp = C + (size_t)(m0 + (half << 3)) * DIM + n0 + l15;
#pragma unroll
    for (int j = 0; j < 8; ++j) {
        cp[(size_t)j * DIM]      = acc0[j];
        cp[(size_t)j * DIM + 16] = acc1[j];
    }
}

// Per-edge degree accumulation: deg_u[row] += val ; deg_i[col] += val
__global__ void deg_kernel(const int* __restrict__ rows, const int* __restrict__ cols,
                           const float* __restrict__ vals,
                           float* __restrict__ degU, float* __restrict__ degI, int E)
{
    int e = blockIdx.x * blockDim.x + threadIdx.x;
    if (e >= E) return;
    float v = vals[e];
    atomicAdd(degU + rows[e], v);
    atomicAdd(degI + cols[e], v);
}

// In-place deg -> 1/(deg+1)
__global__ void div_kernel(float* __restrict__ deg, int n)
{
    int i = blockIdx.x * blockDim.x + threadIdx.x;
    if (i < n) deg[i] = 1.0f / (deg[i] + 1.0f);
}

// One wave per edge: dst[d] += val * div[d] * T[s]  (128 floats, 4 per lane)
__global__ __launch_bounds__(256) void scatter_kernel(
    const int* __restrict__ srcIdx, const int* __restrict__ dstIdx,
    const float* __restrict__ vals, const float* __restrict__ divv,
    const float* __restrict__ T, float* __restrict__ dst, int E)
{
    const int wave = threadIdx.x >> 5;
    const int lane = threadIdx.x & 31;
    int e = blockIdx.x * 8 + wave;
    if (e >= E) return;
    int s = srcIdx[e];
    int d = dstIdx[e];
    float scale = vals[e] * divv[d];
    const float4 v = ((const float4*)(T + (size_t)s * DIM))[lane];
    float* outp = dst + (size_t)d * DIM + lane * 4;
    atomicAdd(outp + 0, v.x * scale);
    atomicAdd(outp + 1, v.y * scale);
    atomicAdd(outp + 2, v.z * scale);
    atomicAdd(outp + 3, v.w * scale);
}

__global__ void leaky_kernel(float* __restrict__ x, long long n)
{
    long long i = (long long)blockIdx.x * blockDim.x + threadIdx.x;
    if (i < n) {
        float v = x[i];
        x[i] = v > 0.0f ? v : 0.01f * v;
    }
}

extern "C" void kernel_launch(void* const* d_in, const int* in_sizes, int n_in,
                              void* d_out, int out_size, void* d_ws, size_t ws_size,
                              hipStream_t stream)
{
    const float* uEmb = (const float*)d_in[0];
    const float* iEmb = (const float*)d_in[1];
    const float* Wp   = (const float*)d_in[2];
    const float* Ws   = (const float*)d_in[3];
    const float* vals = (const float*)d_in[4];
    const int*   rows = (const int*)d_in[5];
    const int*   cols = (const int*)d_in[6];
    float*       out  = (float*)d_out;

    const int NU = in_sizes[0] / DIM;
    const int NI = in_sizes[1] / DIM;
    const int R  = in_sizes[3] / (DIM * DIM);
    const int E  = in_sizes[4] / R;
    const int NMAX = NU > NI ? NU : NI;

    // Workspace layout: u ping buffer, i ping buffer, transformed feats T, degrees
    float* uA   = (float*)d_ws;
    float* iA   = uA + (size_t)NU * DIM;
    float* T    = iA + (size_t)NI * DIM;
    float* degU = T  + (size_t)NMAX * DIM;
    float* degI = degU + NU;

    float* uOut = out;
    float* iOut = out + (size_t)NU * DIM;

    const float* uCur = uEmb;
    const float* iCur = iEmb;

    for (int r = 0; r < R; ++r) {
        // r even -> ws buffers, r odd -> d_out halves; for R=4 the final
        // iteration lands in d_out.
        float* uNxt = (r & 1) ? uOut : uA;
        float* iNxt = (r & 1) ? iOut : iA;
        const int*   ro = rows + (size_t)r * E;
        const int*   co = cols + (size_t)r * E;
        const float* va = vals + (size_t)r * E;
        const float* Wr = Ws + (size_t)r * DIM * DIM;

        // degrees and 1/(deg+1)
        hipMemsetAsync(degU, 0, sizeof(float) * (size_t)(NU + NI), stream);
        deg_kernel<<<(E + 255) / 256, 256, 0, stream>>>(ro, co, va, degU, degI, E);
        div_kernel<<<(NU + NI + 255) / 256, 256, 0, stream>>>(degU, NU + NI);

        // u update: u' = u @ Wp + scatter(val * div_u[row] * (i @ Wr)[col])
        wmma_gemm_f32<<<NI / 16, 128, 0, stream>>>(iCur, Wr, T, NI);
        wmma_gemm_f32<<<NU / 16, 128, 0, stream>>>(uCur, Wp, uNxt, NU);
        scatter_kernel<<<(E + 7) / 8, 256, 0, stream>>>(co, ro, va, degU, T, uNxt, E);

        // i update: i' = i @ Wp + scatter(val * div_i[col] * (u' @ Wr)[row])
        wmma_gemm_f32<<<NU / 16, 128, 0, stream>>>(uNxt, Wr, T, NU);
        wmma_gemm_f32<<<NI / 16, 128, 0, stream>>>(iCur, Wp, iNxt, NI);
        scatter_kernel<<<(E + 7) / 8, 256, 0, stream>>>(ro, co, va, degI, T, iNxt, E);

        uCur = uNxt;
        iCur = iNxt;
    }

    // If R is odd (not the case for R=4), move final buffers into d_out.
    if (uCur != uOut)
        hipMemcpyAsync(uOut, uCur, sizeof(float) * (size_t)NU * DIM,
                       hipMemcpyDeviceToDevice, stream);
    if (iCur != iOut)
        hipMemcpyAsync(iOut, iCur, sizeof(float) * (size_t)NI * DIM,
                       hipMemcpyDeviceToDevice, stream);

    long long total = (long long)(NU + NI) * DIM;
    leaky_kernel<<<(int)((total + 255) / 256), 256, 0, stream>>>(out, total);
}